// KNet_RNN_91104846282818
// MI455X (gfx1250) — compile-verified
//
#include <hip/hip_runtime.h>
#include <hip/hip_bf16.h>

typedef float v2f __attribute__((ext_vector_type(2)));
typedef float v8f __attribute__((ext_vector_type(8)));

#define B_DIM 16
#define M_DIM 15
#define H_DIM 225
#define T_STEPS 256
#define SPLITK 64
#define K2B 18000
#define KC2B 284                              // multiple of 4; 63*284=17892, last split = 108
#define NT2B 15                               // ceil(225/16)

__device__ __forceinline__ v8f wmma_f32(v2f a, v2f b, v8f c) {
    return __builtin_amdgcn_wmma_f32_16x16x4_f32(false, a, false, b, (short)0, c,
                                                 false, false);
}

// Guarded loader for X = concat(X0[16xK0], X1[16xK1]) row-major, zero past Klim.
__device__ __forceinline__ float ldx(const float* __restrict__ X0, int K0,
                                     const float* __restrict__ X1, int K1,
                                     int m, int k, int Klim) {
    if (k >= Klim) return 0.0f;
    if (k < K0) return X0[m * K0 + k];
    return X1[m * K1 + (k - K0)];
}

// One wave (32 threads) computes a 16x16 tile of Y = act(X @ W^T + b).
// X = concat(X0,X1) is (16 x K) row-major, W is (Nout x K) row-major.
// K loop is segmented so the hot chunks have NO per-element guards:
//   [fast X0 chunks] [guarded straddle] [fast X1 chunks] [guarded tail]
template <int RELU>
__global__ void linear_wmma(const float* __restrict__ X0, int K0,
                            const float* __restrict__ X1, int K1,
                            const float* __restrict__ W,
                            const float* __restrict__ bias,
                            float* __restrict__ Y, int Nout) {
    const int K    = K0 + K1;
    const int lane = threadIdx.x & 31;
    const int half = lane >> 4;     // 0: K pair {0,1}; 1: K pair {2,3}
    const int col  = lane & 15;     // A: row M; B: column N
    const int ncol = blockIdx.x * 16 + col;
    const bool nok = (ncol < Nout);
    // out-of-range lanes read W row 0: their B-column only feeds their own
    // (never stored) D-column, so unguarded loads are safe.
    const float* __restrict__ Wrow = W + (size_t)(nok ? ncol : 0) * K;
    const float* __restrict__ Xr0  = X0 + col * K0;

    v8f acc = {};
    int k0 = 0;

    // fast chunks entirely inside X0
#pragma unroll 2
    for (; k0 + 4 <= K0; k0 += 4) {
        const int ka = k0 + half * 2;
        v2f a, b;
        a.x = Xr0[ka];  a.y = Xr0[ka + 1];
        b.x = Wrow[ka]; b.y = Wrow[ka + 1];
        acc = wmma_f32(a, b, acc);
    }
    // straddle chunk: X0 tail and/or X0->X1 boundary (guards run once)
    if (k0 < K0) {
        const int ka = k0 + half * 2;
        v2f a, b;
        a.x = ldx(X0, K0, X1, K1, col, ka,     K);
        a.y = ldx(X0, K0, X1, K1, col, ka + 1, K);
        b.x = (ka     < K) ? Wrow[ka]     : 0.0f;
        b.y = (ka + 1 < K) ? Wrow[ka + 1] : 0.0f;
        acc = wmma_f32(a, b, acc);
        k0 += 4;
    }
    // fast chunks entirely inside X1
    if (K1 > 0) {
        const float* __restrict__ Xr1 = X1 + col * K1;
#pragma unroll 2
        for (; k0 + 4 <= K; k0 += 4) {
            const int ka = k0 + half * 2;
            v2f a, b;
            a.x = Xr1[ka - K0];  a.y = Xr1[ka + 1 - K0];
            b.x = Wrow[ka];      b.y = Wrow[ka + 1];
            acc = wmma_f32(a, b, acc);
        }
    }
    // tail chunk (guards run once)
    if (k0 < K) {
        const int ka = k0 + half * 2;
        v2f a, b;
        a.x = ldx(X0, K0, X1, K1, col, ka,     K);
        a.y = ldx(X0, K0, X1, K1, col, ka + 1, K);
        b.x = (ka     < K) ? Wrow[ka]     : 0.0f;
        b.y = (ka + 1 < K) ? Wrow[ka + 1] : 0.0f;
        acc = wmma_f32(a, b, acc);
    }

    if (!nok) return;
    const float bv = bias[ncol];
#pragma unroll
    for (int r = 0; r < 8; ++r) {
        const int m = r + half * 8;           // C/D layout: VGPR r -> M = r (+8 hi half)
        float v = acc[r] + bv;
        if (RELU) v = fmaxf(v, 0.0f);
        Y[m * Nout + ncol] = v;
    }
}

// Split-K GEMM partial for W2b (K=18000). KC2B and K are multiples of 4, so the
// loop is fully unguarded and both operands use 8-byte vector loads.
__global__ void linear_splitk(const float* __restrict__ X, int K,
                              const float* __restrict__ W, int Nout,
                              int nTiles, float* __restrict__ partials) {
    const int lane  = threadIdx.x & 31;
    const int half  = lane >> 4;
    const int col   = lane & 15;
    const int tile  = blockIdx.x % nTiles;
    const int split = blockIdx.x / nTiles;
    const int ncol  = tile * 16 + col;
    const bool nok  = (ncol < Nout);
    const float* __restrict__ Wrow = W + (size_t)(nok ? ncol : 0) * K;
    const float* __restrict__ Xrow = X + (size_t)col * K;
    const int kbeg = split * KC2B;
    const int kend = (kbeg + KC2B < K) ? (kbeg + KC2B) : K;

    v8f acc = {};
#pragma unroll 4
    for (int k0 = kbeg; k0 < kend; k0 += 4) {
        const int ka = k0 + half * 2;       // even; rows are 4-float aligned
        const v2f a = *(const v2f*)(Xrow + ka);
        const v2f b = *(const v2f*)(Wrow + ka);
        acc = wmma_f32(a, b, acc);
    }
    float* out = partials + ((size_t)split * nTiles + tile) * 256;
#pragma unroll
    for (int r = 0; r < 8; ++r)
        out[(r + half * 8) * 16 + col] = acc[r];
}

__global__ void reduce_out2(const float* __restrict__ partials,
                            const float* __restrict__ bias,
                            float* __restrict__ Y, int Nout, int nTiles) {
    const int idx = blockIdx.x * blockDim.x + threadIdx.x;
    if (idx >= B_DIM * Nout) return;
    const int b = idx / Nout, n = idx % Nout;
    const int tile = n >> 4, c = n & 15;
    float s = bias[n];
#pragma unroll 4
    for (int sp = 0; sp < SPLITK; ++sp)
        s += partials[((size_t)sp * nTiles + tile) * 256 + b * 16 + c];
    Y[idx] = s;
}

// r = sig(gi_r+gh_r); z = sig(gi_z+gh_z); n = tanh(gi_n + r*gh_n); h' = (1-z)n + z h
__global__ void gru_fuse(const float* __restrict__ gi, const float* __restrict__ gh,
                         const float* __restrict__ h, float* __restrict__ hout) {
    const int idx = blockIdx.x * blockDim.x + threadIdx.x;
    if (idx >= B_DIM * H_DIM) return;
    const int b = idx / H_DIM, j = idx % H_DIM;
    const float* gib = gi + b * 3 * H_DIM;
    const float* ghb = gh + b * 3 * H_DIM;
    const float r = 1.0f / (1.0f + __expf(-(gib[j] + ghb[j])));
    const float z = 1.0f / (1.0f + __expf(-(gib[H_DIM + j] + ghb[H_DIM + j])));
    const float n = tanhf(gib[2 * H_DIM + j] + r * ghb[2 * H_DIM + j]);
    hout[idx] = (1.0f - z) * n + z * h[idx];
}

__global__ void prep(const float* __restrict__ y_t, const float* __restrict__ x_post,
                     const float* __restrict__ x_post_prev,
                     const float* __restrict__ x_prior_prev,
                     const float* __restrict__ y_prev,
                     float* __restrict__ in30, float* __restrict__ fw_evol,
                     float* __restrict__ fw_upd) {
    const int idx = threadIdx.x;
    if (idx >= B_DIM * M_DIM) return;
    const int b = idx / M_DIM, i = idx % M_DIM;
    const float obs = y_t[b * 30 + i];
    const float xp  = x_post[idx];
    in30[b * 30 + i]          = obs - y_prev[idx];       // obs_diff
    in30[b * 30 + M_DIM + i]  = obs - xp;                // obs_innov_diff
    fw_evol[idx] = xp - x_post_prev[idx];
    fw_upd[idx]  = xp - x_prior_prev[idx];
}

__global__ void final_update(const float* __restrict__ y_t, const float* __restrict__ out2,
                             float* __restrict__ x_post, float* __restrict__ x_post_prev,
                             float* __restrict__ x_prior_prev, float* __restrict__ y_prev,
                             float* __restrict__ out_t) {
    const int idx = threadIdx.x;
    if (idx >= B_DIM * M_DIM) return;
    const int b = idx / M_DIM, i = idx % M_DIM;
    const float obs = y_t[b * 30 + i];
    const float xp  = x_post[idx];
    const float dy  = obs - xp;
    const float kg  = out2[b * H_DIM + i];   // KG[:,0,:] = first 15 of each row
    const float xn  = xp + kg * dy;
    x_post_prev[idx]  = xp;
    x_prior_prev[idx] = xp;                  // x_prior == old x_post
    y_prev[idx]       = obs;
    x_post[idx]       = xn;
    out_t[idx]        = xn;
}

__global__ void init_state(const float* __restrict__ x0, const float* __restrict__ hQ0,
                           const float* __restrict__ hSig0, const float* __restrict__ hS0,
                           float* x_post, float* x_post_prev, float* x_prior_prev,
                           float* y_prev, float* hQ, float* hSig, float* hS) {
    const int idx = blockIdx.x * blockDim.x + threadIdx.x;
    if (idx < B_DIM * M_DIM) {
        const float v = x0[idx];
        x_post[idx] = v; x_post_prev[idx] = v; x_prior_prev[idx] = v; y_prev[idx] = v;
    }
    if (idx < B_DIM * H_DIM) {
        hQ[idx] = hQ0[idx]; hSig[idx] = hSig0[idx]; hS[idx] = hS0[idx];
    }
}

extern "C" void kernel_launch(void* const* d_in, const int* in_sizes, int n_in,
                              void* d_out, int out_size, void* d_ws, size_t ws_size,
                              hipStream_t stream) {
    const float* y      = (const float*)d_in[0];
    const float* x0     = (const float*)d_in[1];
    const float* hQ0    = (const float*)d_in[2];
    const float* hSig0  = (const float*)d_in[3];
    const float* hS0    = (const float*)d_in[4];
    const float* WihQ   = (const float*)d_in[5];
    const float* WhhQ   = (const float*)d_in[6];
    const float* bihQ   = (const float*)d_in[7];
    const float* bhhQ   = (const float*)d_in[8];
    const float* WihSig = (const float*)d_in[9];
    const float* WhhSig = (const float*)d_in[10];
    const float* bihSig = (const float*)d_in[11];
    const float* bhhSig = (const float*)d_in[12];
    const float* WihS   = (const float*)d_in[13];
    const float* WhhS   = (const float*)d_in[14];
    const float* bihS   = (const float*)d_in[15];
    const float* bhhS   = (const float*)d_in[16];
    const float* W1  = (const float*)d_in[17]; const float* b1  = (const float*)d_in[18];
    const float* W2a = (const float*)d_in[19]; const float* b2a = (const float*)d_in[20];
    const float* W2b = (const float*)d_in[21]; const float* b2b = (const float*)d_in[22];
    const float* W3  = (const float*)d_in[23]; const float* b3  = (const float*)d_in[24];
    const float* W4  = (const float*)d_in[25]; const float* b4  = (const float*)d_in[26];
    const float* W5  = (const float*)d_in[27]; const float* b5  = (const float*)d_in[28];
    const float* W6  = (const float*)d_in[29]; const float* b6  = (const float*)d_in[30];
    const float* W7  = (const float*)d_in[31]; const float* b7  = (const float*)d_in[32];
    float* out = (float*)d_out;

    // workspace layout (floats)
    float* w = (float*)d_ws;
    float* x_post       = w;            // 240
    float* x_post_prev  = w + 240;      // 240
    float* x_prior_prev = w + 480;      // 240
    float* y_prev       = w + 720;      // 240
    float* hQ           = w + 960;      // 3600
    float* hSig         = w + 4560;     // 3600
    float* hS           = w + 8160;     // 3600
    float* in30         = w + 11760;    // 480
    float* fw_evol      = w + 12240;    // 240
    float* fw_upd       = w + 12480;    // 240
    float* out5         = w + 12720;    // 1200
    float* out6         = w + 13920;    // 1200
    float* out7         = w + 15120;    // 2400
    float* giQ          = w + 17520;    // 10800
    float* ghQ          = w + 28320;    // 10800
    float* giSig        = w + 39120;    // 10800
    float* ghSig        = w + 49920;    // 10800
    float* giS          = w + 60720;    // 10800
    float* ghS          = w + 71520;    // 10800
    float* out_sigma    = w + 82320;    // 3600
    float* out1         = w + 85920;    // 3600
    float* h2           = w + 89520;    // 16*18000 = 288000
    float* out2         = w + 377520;   // 3600
    float* out3         = w + 381120;   // 3600
    float* partials     = w + 384720;   // SPLITK*NT2B*256 = 245760

    const int gH   = (B_DIM * H_DIM + 255) / 256;  // 15 blocks of 256 over 16x225
    const int g675 = (675 + 15) / 16;              // 43 tiles
    const int g225 = (225 + 15) / 16;              // 15 tiles
    const int g75  = (75 + 15) / 16;               // 5 tiles
    const int g150 = (150 + 15) / 16;              // 10 tiles
    const int g18k = K2B / 16;                     // 1125 tiles

    init_state<<<gH, 256, 0, stream>>>(x0, hQ0, hSig0, hS0, x_post, x_post_prev,
                                       x_prior_prev, y_prev, hQ, hSig, hS);

    for (int t = 0; t < T_STEPS; ++t) {
        const float* y_t = y + (size_t)t * B_DIM * 30;

        prep<<<1, 256, 0, stream>>>(y_t, x_post, x_post_prev, x_prior_prev, y_prev,
                                    in30, fw_evol, fw_upd);

        // GRU Q
        linear_wmma<1><<<g75, 32, 0, stream>>>(fw_evol, M_DIM, nullptr, 0, W5, b5, out5, 75);
        linear_wmma<0><<<g675, 32, 0, stream>>>(out5, 75, nullptr, 0, WihQ, bihQ, giQ, 675);
        linear_wmma<0><<<g675, 32, 0, stream>>>(hQ, H_DIM, nullptr, 0, WhhQ, bhhQ, ghQ, 675);
        gru_fuse<<<gH, 256, 0, stream>>>(giQ, ghQ, hQ, hQ);

        // GRU Sigma (input = concat(hQ, out6)); output kept as out_sigma
        linear_wmma<1><<<g75, 32, 0, stream>>>(fw_upd, M_DIM, nullptr, 0, W6, b6, out6, 75);
        linear_wmma<0><<<g675, 32, 0, stream>>>(hQ, H_DIM, out6, 75, WihSig, bihSig, giSig, 675);
        linear_wmma<0><<<g675, 32, 0, stream>>>(hSig, H_DIM, nullptr, 0, WhhSig, bhhSig, ghSig, 675);
        gru_fuse<<<gH, 256, 0, stream>>>(giSig, ghSig, hSig, out_sigma);

        // GRU S (input = concat(out1, out7))
        linear_wmma<1><<<g225, 32, 0, stream>>>(out_sigma, H_DIM, nullptr, 0, W1, b1, out1, 225);
        linear_wmma<1><<<g150, 32, 0, stream>>>(in30, 30, nullptr, 0, W7, b7, out7, 150);
        linear_wmma<0><<<g675, 32, 0, stream>>>(out1, 225, out7, 150, WihS, bihS, giS, 675);
        linear_wmma<0><<<g675, 32, 0, stream>>>(hS, H_DIM, nullptr, 0, WhhS, bhhS, ghS, 675);
        gru_fuse<<<gH, 256, 0, stream>>>(giS, ghS, hS, hS);

        // out2 = (relu(concat(out_sigma,hS) @ W2a^T + b2a)) @ W2b^T + b2b
        linear_wmma<1><<<g18k, 32, 0, stream>>>(out_sigma, H_DIM, hS, H_DIM, W2a, b2a, h2, K2B);
        linear_splitk<<<NT2B * SPLITK, 32, 0, stream>>>(h2, K2B, W2b, 225, NT2B, partials);
        reduce_out2<<<gH, 256, 0, stream>>>(partials, b2b, out2, 225, NT2B);

        // out3, out4 (out4 becomes the new hSig carry)
        linear_wmma<1><<<g225, 32, 0, stream>>>(hS, H_DIM, out2, H_DIM, W3, b3, out3, 225);
        linear_wmma<1><<<g225, 32, 0, stream>>>(out_sigma, H_DIM, out3, H_DIM, W4, b4, hSig, 225);

        final_update<<<1, 256, 0, stream>>>(y_t, out2, x_post, x_post_prev,
                                            x_prior_prev, y_prev,
                                            out + (size_t)t * B_DIM * M_DIM);
    }
}